// ClefAttention_68066641707504
// MI455X (gfx1250) — compile-verified
//
#include <hip/hip_runtime.h>
#include <hip/hip_bf16.h>
#include <math.h>

typedef __attribute__((ext_vector_type(2))) float v2f;
typedef __attribute__((ext_vector_type(8))) float v8f;

// ---------------------------------------------------------------------------
// C[M,N] = A[M,512] * W[N,512]^T + bias[N]      (K fixed = 512)
// One wave32 computes a 16(M) x 64(N) tile via V_WMMA_F32_16X16X4_F32.
// Block = 128 threads = 4 waves stacked along M. Requires M%16==0, N%64==0.
// ---------------------------------------------------------------------------
__global__ __launch_bounds__(128)
void gemm_nt512_wmma(const float* __restrict__ A,
                     const float* __restrict__ W,
                     const float* __restrict__ bias,
                     float* __restrict__ C,
                     int M, int N)
{
    constexpr int K = 512;
    const int lane = threadIdx.x & 31;
    const int wave = threadIdx.x >> 5;
    const int m0   = (blockIdx.y * 4 + wave) * 16;
    const int n0   = blockIdx.x * 64;
    if (m0 >= M) return;                 // wave-uniform exit (EXEC stays all-1s for WMMA)

    const int half = lane >> 4;          // selects K pair within the 16x4 fragment
    const int l16  = lane & 15;

    const float* Ar = A + (size_t)(m0 + l16) * K;
    const float* W0 = W + (size_t)(n0 + l16) * K;
    const float* W1 = W0 + (size_t)16 * K;
    const float* W2 = W0 + (size_t)32 * K;
    const float* W3 = W0 + (size_t)48 * K;

    __builtin_prefetch(Ar, 0, 3);        // global_prefetch_b8
    __builtin_prefetch(W0, 0, 3);

    v8f acc0 = {}, acc1 = {}, acc2 = {}, acc3 = {};

    #pragma unroll 4
    for (int k = 0; k < K; k += 4) {
        const int kb = k + 2 * half;
        // A fragment: 16x4 fp32 (lanes 0-15 -> K={k,k+1}, lanes 16-31 -> K={k+2,k+3})
        v2f av = *(const v2f*)(Ar + kb);
        // B fragment: 4x16 fp32 of W^T, i.e. W rows striped over lanes (N), K over half
        v2f b0 = *(const v2f*)(W0 + kb);
        v2f b1 = *(const v2f*)(W1 + kb);
        v2f b2 = *(const v2f*)(W2 + kb);
        v2f b3 = *(const v2f*)(W3 + kb);
        acc0 = __builtin_amdgcn_wmma_f32_16x16x4_f32(false, av, false, b0, (short)0, acc0, false, false);
        acc1 = __builtin_amdgcn_wmma_f32_16x16x4_f32(false, av, false, b1, (short)0, acc1, false, false);
        acc2 = __builtin_amdgcn_wmma_f32_16x16x4_f32(false, av, false, b2, (short)0, acc2, false, false);
        acc3 = __builtin_amdgcn_wmma_f32_16x16x4_f32(false, av, false, b3, (short)0, acc3, false, false);
    }

    // Epilogue: C/D layout -> VGPR r holds M=r (lanes 0-15) / M=r+8 (lanes 16-31), N=lane%16.
    const int mb = m0 + half * 8;
    const float bv0 = bias[n0 +  0 + l16];
    const float bv1 = bias[n0 + 16 + l16];
    const float bv2 = bias[n0 + 32 + l16];
    const float bv3 = bias[n0 + 48 + l16];
    #pragma unroll
    for (int r = 0; r < 8; ++r) {
        float* Crow = C + (size_t)(mb + r) * N + n0 + l16;
        Crow[ 0] = acc0[r] + bv0;
        Crow[16] = acc1[r] + bv1;
        Crow[32] = acc2[r] + bv2;
        Crow[48] = acc3[r] + bv3;
    }
}

// ---------------------------------------------------------------------------
// Sampling: one block per (b,q); wave = head; lane owns 2 of Dh=64 channels.
// Computes tanh offsets, softmax over L*K=16, bilinear gather from vp, blend.
//   tl/fl : (B*Nq, 64)  time/freq offset logits
//   al    : (B*Nq, 128) attention logits
//   vp    : (B*Nv, 512) projected values, viewed as (B,Nv,H=8,Dh=64)
//   samp  : (B*Nq, 512) output, viewed as (B,Nq,H,Dh)
// ---------------------------------------------------------------------------
__global__ __launch_bounds__(256)
void clef_sample_kernel(const float* __restrict__ tl,
                        const float* __restrict__ fl,
                        const float* __restrict__ al,
                        const float* __restrict__ rp,   // (B,Nq,L,2)
                        const float* __restrict__ vr,   // (B,L,2)
                        const int*   __restrict__ ss,   // (L,2) = (H_l, W_l)
                        const int*   __restrict__ lsi,  // (L)
                        const float* __restrict__ vp,
                        float* __restrict__ samp,
                        int B, int Nq, int Nv)
{
    const int bq   = blockIdx.x;
    const int b    = bq / Nq;
    const int h    = threadIdx.x >> 5;   // head = wave id (0..7)
    const int lane = threadIdx.x & 31;
    const int c0   = lane * 2;

    // softmax over the 16 (L*K) logits of this head
    const float* a = al + (size_t)bq * 128 + h * 16;
    float lg[16];
    float mx = -1e30f;
    #pragma unroll
    for (int j = 0; j < 16; ++j) { lg[j] = a[j]; mx = fmaxf(mx, lg[j]); }
    float sum = 0.f;
    #pragma unroll
    for (int j = 0; j < 16; ++j) { lg[j] = __expf(lg[j] - mx); sum += lg[j]; }
    const float inv = 1.f / sum;

    const float* tb = tl + (size_t)bq * 64 + h * 8;   // (L,Kt) for this head
    const float* fb = fl + (size_t)bq * 64 + h * 8;   // (L,Kf)

    float acc0 = 0.f, acc1 = 0.f;

    #pragma unroll
    for (int l = 0; l < 4; ++l) {
        const int   Hl = ss[2 * l], Wl = ss[2 * l + 1];
        const int   start = lsi[l];
        const float rx  = rp[((size_t)bq * 4 + l) * 2 + 0];
        const float ry  = rp[((size_t)bq * 4 + l) * 2 + 1];
        const float vrx = vr[(b * 4 + l) * 2 + 0];
        const float vry = vr[(b * 4 + l) * 2 + 1];
        float to[2], fo[2];
        #pragma unroll
        for (int i = 0; i < 2; ++i) {
            to[i] = tanhf(tb[l * 2 + i]) * 0.15f;
            fo[i] = tanhf(fb[l * 2 + i]) * 0.15f;
        }
        #pragma unroll
        for (int kt = 0; kt < 2; ++kt) {
            #pragma unroll
            for (int kf = 0; kf < 2; ++kf) {
                float lx = (rx + to[kt] / (float)Wl) * vrx;   // norm = (W,H)
                float ly = (ry + fo[kf] / (float)Hl) * vry;
                lx = fminf(fmaxf(lx, 0.f), 1.f);
                ly = fminf(fmaxf(ly, 0.f), 1.f);
                // grid_sample align_corners=False: ix = loc*W - 0.5
                const float ix = lx * (float)Wl - 0.5f;
                const float iy = ly * (float)Hl - 0.5f;
                const float x0 = floorf(ix), y0 = floorf(iy);
                const float aw = lg[l * 4 + kt * 2 + kf] * inv;
                #pragma unroll
                for (int dx = 0; dx < 2; ++dx) {
                    #pragma unroll
                    for (int dy = 0; dy < 2; ++dy) {
                        const float xc = x0 + (float)dx, yc = y0 + (float)dy;
                        if (xc >= 0.f && xc < (float)Wl && yc >= 0.f && yc < (float)Hl) {
                            const float w = (1.f - fabsf(ix - xc)) *
                                            (1.f - fabsf(iy - yc)) * aw;
                            const int xi = (int)xc, yi = (int)yc;
                            const size_t row = (size_t)(b * Nv + start + yi * Wl + xi);
                            const float2 v = *(const float2*)(vp + (row * 8 + h) * 64 + c0);
                            acc0 += w * v.x;
                            acc1 += w * v.y;
                        }
                    }
                }
            }
        }
    }
    float2 o; o.x = acc0; o.y = acc1;
    *(float2*)(samp + ((size_t)bq * 8 + h) * 64 + c0) = o;
}

// ---------------------------------------------------------------------------
extern "C" void kernel_launch(void* const* d_in, const int* in_sizes, int n_in,
                              void* d_out, int out_size, void* d_ws, size_t ws_size,
                              hipStream_t stream)
{
    const float* query   = (const float*)d_in[0];
    const float* refp    = (const float*)d_in[1];
    const float* value   = (const float*)d_in[2];
    const int*   ss      = (const int*)  d_in[3];
    const int*   lsi     = (const int*)  d_in[4];
    const float* vratio  = (const float*)d_in[5];
    const float* time_w  = (const float*)d_in[6];
    const float* time_b  = (const float*)d_in[7];
    const float* freq_w  = (const float*)d_in[8];
    const float* freq_b  = (const float*)d_in[9];
    const float* attn_w  = (const float*)d_in[10];
    const float* attn_b  = (const float*)d_in[11];
    const float* value_w = (const float*)d_in[12];
    const float* value_b = (const float*)d_in[13];
    const float* out_w   = (const float*)d_in[14];
    const float* out_b   = (const float*)d_in[15];

    const int D   = 512;
    const int L   = 4;
    const int BNq = in_sizes[0] / D;        // B*Nq
    const int BNv = in_sizes[2] / D;        // B*Nv
    const int B   = in_sizes[5] / (L * 2);
    const int Nq  = BNq / B;
    const int Nv  = BNv / B;

    // workspace partition (floats)
    float* ws   = (float*)d_ws;
    float* vp   = ws;                               // BNv*512
    float* tlog = vp   + (size_t)BNv * 512;         // BNq*64
    float* flog = tlog + (size_t)BNq * 64;          // BNq*64
    float* alog = flog + (size_t)BNq * 64;          // BNq*128
    float* samp = alog + (size_t)BNq * 128;         // BNq*512
    (void)ws_size; (void)n_in; (void)out_size;

    const dim3 blk(128);

    // 1) value projection: vp = value @ value_w^T + value_b
    gemm_nt512_wmma<<<dim3(512 / 64, (BNv + 63) / 64), blk, 0, stream>>>(
        value, value_w, value_b, vp, BNv, 512);

    // 2-4) query projections
    gemm_nt512_wmma<<<dim3(64 / 64, (BNq + 63) / 64), blk, 0, stream>>>(
        query, time_w, time_b, tlog, BNq, 64);
    gemm_nt512_wmma<<<dim3(64 / 64, (BNq + 63) / 64), blk, 0, stream>>>(
        query, freq_w, freq_b, flog, BNq, 64);
    gemm_nt512_wmma<<<dim3(128 / 64, (BNq + 63) / 64), blk, 0, stream>>>(
        query, attn_w, attn_b, alog, BNq, 128);

    // 5) deformable bilinear sampling + attention blend
    clef_sample_kernel<<<dim3(BNq), dim3(256), 0, stream>>>(
        tlog, flog, alog, refp, vratio, ss, lsi, vp, samp, B, Nq, Nv);

    // 6) output projection: out = samp @ out_w^T + out_b
    gemm_nt512_wmma<<<dim3(512 / 64, (BNq + 63) / 64), blk, 0, stream>>>(
        samp, out_w, out_b, (float*)d_out, BNq, 512);
}